// PSAttentionMasked_46050639347858
// MI455X (gfx1250) — compile-verified
//
#include <hip/hip_runtime.h>
#include <stdint.h>

#define HH 256
#define WW 256
#define HWH (HH * WW)
#define CC 32
#define KK 4
#define NE 36 /* 9 patch offsets * K */

// ---------------------------------------------------------------------------
// Kernel 1: transpose v (B,C,H,W) -> vT (B, H*W, C) so each pixel's 32
// channels are one contiguous 128B line for the gather phase.
// Uses CDNA5 async global->LDS (b128) to stage the tile, then coalesced
// 128B-per-pixel stores.
// ---------------------------------------------------------------------------
__global__ __launch_bounds__(256) void transpose_v_kernel(
    const float* __restrict__ v, float* __restrict__ vT) {
  // padded row (36 floats = 144B, still 16B-aligned chunks) -> only 2-way
  // bank conflict on the column read instead of 32-way.
  __shared__ __align__(16) float tile[CC][36];

  const int blk = blockIdx.x;          // B * (HW/32) = 4096 blocks
  const int b = blk >> 11;             // HW/32 = 2048 per batch
  const int pixbase = (blk & 2047) * 32;
  const int t = threadIdx.x;

  // Async stage: each thread moves 16B (4 pixels of one channel) to LDS.
  {
    const int c = t >> 3;            // 0..31
    const int jp4 = (t & 7) * 4;     // 0,4,...,28
    const uint64_t gaddr =
        (uint64_t)(uintptr_t)(v + ((size_t)(b * CC + c) * HWH + pixbase + jp4));
    const uint32_t ldsoff = (uint32_t)(uintptr_t)(&tile[c][jp4]);
    asm volatile("global_load_async_to_lds_b128 %0, %1, off"
                 :: "v"(ldsoff), "v"(gaddr)
                 : "memory");
  }
  asm volatile("s_wait_asynccnt 0" ::: "memory");
  __syncthreads();

  // Write out: 32 consecutive channels of one pixel = 128B coalesced store.
  {
    const int c = t & 31;
    const int p0 = t >> 5;           // 0..7
    float* dst = vT + ((size_t)b * HWH + pixbase) * CC + c;
#pragma unroll
    for (int r = 0; r < 4; ++r) {
      const int pix = p0 + r * 8;
      dst[(size_t)pix * CC] = tile[c][pix];
    }
  }
}

// ---------------------------------------------------------------------------
// Kernel 2: softmax weights + gather + weighted sum.
// Block = 256 threads = 64 consecutive flattened pixels of one batch.
// Phase 1: 256 threads compute all 36x64 (weight, gather-addr) pairs -> LDS.
// Phase 2: 4 threads per pixel, 8 channels each; normalize by running wsum.
// ---------------------------------------------------------------------------
template <bool USE_VT>
__global__ __launch_bounds__(256) void ps_attn_kernel(
    const float* __restrict__ v,      // (B,C,H,W)
    const float* __restrict__ cost,   // (B,K,H,W)
    const int* __restrict__ shift,    // (B,2,K,H,W)
    const float* __restrict__ vT,     // (B,HW,C) or null
    float* __restrict__ out) {        // (B,C,H,W)
  __shared__ float lds_w[NE][64];
  __shared__ int lds_a[NE][64];

  const int blk = blockIdx.x;          // B * (HW/64) = 2048 blocks
  const int b = blk >> 10;             // HW/64 = 1024 per batch
  const int pixbase = (blk & 1023) * 64;
  const int t = threadIdx.x;

  // ---- Phase 1: weights + gather addresses (coalesced cost/shift loads) ----
  {
    const int pixl = t & 63;
    const int pix = pixbase + pixl;
    const int i = pix >> 8;            // / W
    const int j = pix & 255;           // % W
    const int e0 = t >> 6;             // 0..3
#pragma unroll
    for (int r = 0; r < 9; ++r) {
      const int e = e0 + r * 4;        // covers 0..35
      const int k = e & 3;
      const int o = e >> 2;            // 0..8
      const int di = o / 3;
      const int dj = o - di * 3;
      const int si_ = i + di - 1;
      const int sj_ = j + dj - 1;
      const bool inb = ((unsigned)si_ < HH) && ((unsigned)sj_ < WW);
      float cst = 10.0f;
      int s0 = 10, s1 = 10;
      if (inb) {
        const size_t off = ((size_t)si_ << 8) + (size_t)sj_;
        cst = cost[(size_t)(b * KK + k) * HWH + off];
        s0 = shift[(size_t)((b * 2 + 0) * KK + k) * HWH + off];
        s1 = shift[(size_t)((b * 2 + 1) * KK + k) * HWH + off];
      }
      int ii = s0 + (1 - di);
      ii = min(max(ii, 0), HH - 1);
      int jj = s1 + (1 - dj);
      jj = min(max(jj, 0), WW - 1);
      lds_w[e][pixl] = __expf(-cst);          // T = 1.0
      lds_a[e][pixl] = (ii << 8) + jj;
    }
  }
  __syncthreads();

  // ---- Phase 2: gather + weighted accumulate --------------------------------
  {
    const int part = t >> 6;           // 0..3 -> channels part*8 .. part*8+7
    const int pixl = t & 63;           // lanes = consecutive pixels (coalesced)
    const int pix = pixbase + pixl;
    float acc[8];
#pragma unroll
    for (int r = 0; r < 8; ++r) acc[r] = 0.0f;
    float wsum = 0.0f;

#pragma unroll 4
    for (int e = 0; e < NE; ++e) {
      const float w = lds_w[e][pixl];
      const int a = lds_a[e][pixl];
      if (USE_VT) {
        const float* src = vT + ((size_t)b * HWH + (size_t)a) * CC + part * 8;
        const float4 g0 = *(const float4*)(src);
        const float4 g1 = *(const float4*)(src + 4);
        acc[0] += w * g0.x; acc[1] += w * g0.y;
        acc[2] += w * g0.z; acc[3] += w * g0.w;
        acc[4] += w * g1.x; acc[5] += w * g1.y;
        acc[6] += w * g1.z; acc[7] += w * g1.w;
      } else {
        const float* sv = v + (size_t)b * CC * HWH + (size_t)a;
#pragma unroll
        for (int r = 0; r < 8; ++r)
          acc[r] += w * sv[(size_t)(part * 8 + r) * HWH];
      }
      wsum += w;
    }

    const float inv = 1.0f / wsum;     // softmax normalization folded in
#pragma unroll
    for (int r = 0; r < 8; ++r) {      // fixed r: 32 lanes = 128B coalesced
      const int c = part * 8 + r;
      out[(size_t)(b * CC + c) * HWH + pix] = acc[r] * inv;
    }
  }
}

// ---------------------------------------------------------------------------
extern "C" void kernel_launch(void* const* d_in, const int* in_sizes, int n_in,
                              void* d_out, int out_size, void* d_ws,
                              size_t ws_size, hipStream_t stream) {
  const float* v = (const float*)d_in[0];      // (2,32,256,256) f32
  const float* cost = (const float*)d_in[1];   // (2,4,256,256)  f32
  const int* shift = (const int*)d_in[2];      // (2,2,4,256,256) i32
  float* out = (float*)d_out;                  // (2,32,256,256) f32
  (void)in_sizes; (void)n_in; (void)out_size;

  const int B = 2;
  const size_t vtBytes = (size_t)B * HWH * CC * sizeof(float);

  if (d_ws != nullptr && ws_size >= vtBytes) {
    float* vT = (float*)d_ws;
    transpose_v_kernel<<<B * (HWH / 32), 256, 0, stream>>>(v, vT);
    ps_attn_kernel<true><<<B * (HWH / 64), 256, 0, stream>>>(v, cost, shift,
                                                             vT, out);
  } else {
    ps_attn_kernel<false><<<B * (HWH / 64), 256, 0, stream>>>(
        v, cost, shift, nullptr, out);
  }
}